// WordWeightingLayer_2551210574013
// MI455X (gfx1250) — compile-verified
//
#include <hip/hip_runtime.h>
#include <math.h>

typedef __attribute__((ext_vector_type(2))) float v2f;
typedef __attribute__((ext_vector_type(4))) float v4f;
typedef __attribute__((ext_vector_type(8))) float v8f;

#define DD   256      // D
#define NTOK 16384    // B*S
#define NF   8        // F

// Hardware tanh if the builtin is exposed for gfx1250 (V_TANH_F32 is a CDNA5
// TRANS32 op); otherwise fall back to libm tanhf (ocml expansion).
static __device__ __forceinline__ float hw_tanh(float x) {
#if __has_builtin(__builtin_amdgcn_tanhf)
    return __builtin_amdgcn_tanhf(x);
#elif __has_builtin(__builtin_amdgcn_tanh_f32)
    return __builtin_amdgcn_tanh_f32(x);
#else
    return tanhf(x);
#endif
}

// -------- zero the padding row (table row 0) --------
__global__ __launch_bounds__(256) void zero_row0_kernel(float* __restrict__ table) {
    table[threadIdx.x] = 0.0f;
}

// -------- h = tanh(X @ W1^T + b1)  ->  table rows 1..NTOK --------
// One wave computes a 16(M) x 64(N) output tile with 4 f32 WMMA accumulators.
// V_WMMA_F32_16X16X4_F32: A = 16x4 f32 (2 VGPR/lane), B = 4x16 f32 (2 VGPR/lane),
// C/D = 16x16 f32 (8 VGPR/lane). K loop: 256 / 4 = 64 steps.
__global__ __launch_bounds__(256) void gemm_tanh_kernel(
    const float* __restrict__ X,      // [NTOK, DD]
    const float* __restrict__ W1,     // [DD, DD] row-major (out = X @ W1^T)
    const float* __restrict__ bias1,  // [DD]
    float* __restrict__ table)        // [1+NTOK, DD]
{
    const int lane = threadIdx.x & 31;
    const int wid  = (blockIdx.x << 3) | (threadIdx.x >> 5);  // 8 waves per block
    const int m0   = (wid >> 2) << 4;   // 16-row tile   (1024 m-tiles)
    const int n0   = (wid &  3) << 6;   // 64-col tile   (4 n-tiles)
    const int lo   = lane & 15;
    const int hi   = lane >> 4;         // 0 or 1

    // A frag: row m0+lo, K = k + 2*hi + {0,1}  (contiguous pair -> b64 load)
    const float* aptr = X + (size_t)(m0 + lo) * DD + (hi << 1);
    // B frag j: col n0+16j+lo, same K pair; B[k][n] = W1[n][k]
    const float* bp0 = W1 + (size_t)(n0 +  0 + lo) * DD + (hi << 1);
    const float* bp1 = W1 + (size_t)(n0 + 16 + lo) * DD + (hi << 1);
    const float* bp2 = W1 + (size_t)(n0 + 32 + lo) * DD + (hi << 1);
    const float* bp3 = W1 + (size_t)(n0 + 48 + lo) * DD + (hi << 1);

    v8f c0 = {}, c1 = {}, c2 = {}, c3 = {};

#pragma unroll 2
    for (int k = 0; k < DD; k += 4) {
        v2f a  = *(const v2f*)(aptr + k);
        v2f w0 = *(const v2f*)(bp0 + k);
        v2f w1 = *(const v2f*)(bp1 + k);
        v2f w2 = *(const v2f*)(bp2 + k);
        v2f w3 = *(const v2f*)(bp3 + k);
        c0 = __builtin_amdgcn_wmma_f32_16x16x4_f32(false, a, false, w0, (short)0, c0, false, false);
        c1 = __builtin_amdgcn_wmma_f32_16x16x4_f32(false, a, false, w1, (short)0, c1, false, false);
        c2 = __builtin_amdgcn_wmma_f32_16x16x4_f32(false, a, false, w2, (short)0, c2, false, false);
        c3 = __builtin_amdgcn_wmma_f32_16x16x4_f32(false, a, false, w3, (short)0, c3, false, false);
    }

    const float bb0 = bias1[n0 +  0 + lo];
    const float bb1 = bias1[n0 + 16 + lo];
    const float bb2 = bias1[n0 + 32 + lo];
    const float bb3 = bias1[n0 + 48 + lo];

    // C/D layout: VGPR r holds (row = r + 8*hi, col = lo) of the 16x16 tile.
    float* outb = table + (size_t)(1 + m0 + (hi << 3)) * DD;
#pragma unroll
    for (int r = 0; r < 8; ++r) {
        float* orow = outb + (size_t)r * DD;
        orow[n0 +  0 + lo] = hw_tanh(c0[r] + bb0);
        orow[n0 + 16 + lo] = hw_tanh(c1[r] + bb1);
        orow[n0 + 32 + lo] = hw_tanh(c2[r] + bb2);
        orow[n0 + 48 + lo] = hw_tanh(c3[r] + bb3);
    }
}

// -------- gather 8 rows, rowwise max, dot w2, + b2, + mask --------
// One wave (32 lanes) per token; each lane owns 8 consecutive floats of D.
__global__ __launch_bounds__(256) void gather_max_score_kernel(
    const int*   __restrict__ sidx,   // [NTOK, NF]
    const int*   __restrict__ wmask,  // [NTOK]
    const float* __restrict__ table,  // [1+NTOK, DD]
    const float* __restrict__ w2,     // [DD]
    const float* __restrict__ b2,     // [1]
    float* __restrict__ out)          // [NTOK]
{
    const int lane  = threadIdx.x & 31;
    const int tok   = (blockIdx.x << 3) | (threadIdx.x >> 5);
    const int dbase = lane << 3;

    const int* ip = sidx + (size_t)tok * NF;

    const float* r0 = table + (size_t)ip[0] * DD + dbase;
    v4f g0 = *(const v4f*)(r0);
    v4f g1 = *(const v4f*)(r0 + 4);

#pragma unroll
    for (int f = 1; f < NF; ++f) {
        const float* rf = table + (size_t)ip[f] * DD + dbase;
        v4f a0 = *(const v4f*)(rf);
        v4f a1 = *(const v4f*)(rf + 4);
        g0.x = fmaxf(g0.x, a0.x); g0.y = fmaxf(g0.y, a0.y);
        g0.z = fmaxf(g0.z, a0.z); g0.w = fmaxf(g0.w, a0.w);
        g1.x = fmaxf(g1.x, a1.x); g1.y = fmaxf(g1.y, a1.y);
        g1.z = fmaxf(g1.z, a1.z); g1.w = fmaxf(g1.w, a1.w);
    }

    v4f wv0 = *(const v4f*)(w2 + dbase);
    v4f wv1 = *(const v4f*)(w2 + dbase + 4);
    float s = g0.x * wv0.x + g0.y * wv0.y + g0.z * wv0.z + g0.w * wv0.w
            + g1.x * wv1.x + g1.y * wv1.y + g1.z * wv1.z + g1.w * wv1.w;

    // wave32 butterfly reduction
#pragma unroll
    for (int off = 16; off > 0; off >>= 1)
        s += __shfl_xor(s, off, 32);

    if (lane == 0) {
        float neg = -10000.0f * (1.0f - (float)wmask[tok]);
        out[tok] = s + b2[0] + neg;
    }
}

extern "C" void kernel_launch(void* const* d_in, const int* in_sizes, int n_in,
                              void* d_out, int out_size, void* d_ws, size_t ws_size,
                              hipStream_t stream) {
    const float* hidden = (const float*)d_in[0];  // [B,S,D] f32
    const int*   sidx   = (const int*)  d_in[1];  // [B,S,F] i32
    const int*   wmask  = (const int*)  d_in[2];  // [B,S]   i32
    const float* W1     = (const float*)d_in[3];  // [D,D]   f32
    const float* b1     = (const float*)d_in[4];  // [D]     f32
    const float* w2     = (const float*)d_in[5];  // [D]     f32
    const float* b2     = (const float*)d_in[6];  // [1]     f32
    float* out   = (float*)d_out;                 // [B,S]   f32
    float* table = (float*)d_ws;                  // [1+NTOK, DD] f32 (~16.8 MB)

    // 1) zero padding row 0
    zero_row0_kernel<<<1, 256, 0, stream>>>(table);
    // 2) tanh-GEMM into table rows 1..NTOK: 1024 m-tiles * 4 n-tiles = 4096 waves / 8 per block
    gemm_tanh_kernel<<<512, 256, 0, stream>>>(hidden, W1, b1, table);
    // 3) gather + max + dot: 16384 tokens, 8 waves per block
    gather_max_score_kernel<<<NTOK / 8, 256, 0, stream>>>(sidx, wmask, table, w2, b2, out);
}